// SSIM_38268158607529
// MI455X (gfx1250) — compile-verified
//
#include <hip/hip_runtime.h>
#include <stdint.h>

// SSIM (3x3 SAME box window, count_include_pad=False) on 16x3x512x512 fp32.
// Memory-bound: 151 MB mandatory HBM traffic -> ~6.5us @ 23.3 TB/s.
// Strategy: one fused pass per 64x32 output tile.
//  * Interior blocks: Tensor Data Mover (tensor_load_to_lds) DMAs the whole
//    34x72 strided tile into LDS in ONE instruction per input (TENSORcnt).
//  * Boundary blocks: per-lane global_load_async_to_lds_b128 with zero-fill
//    for SAME padding (ASYNCcnt).
// Then separable 3-tap moment sums in LDS and a coalesced fp32 store.

#define TILE_W    64
#define TILE_H    32
#define IN_ROWS   34                 // TILE_H + 2 halo rows
#define IN_F4     18                 // float4s per staged row (72 cols, 16B-aligned)
#define IN_STRIDE 72                 // floats per staged LDS row
#define IN_SLOTS4 (IN_ROWS * IN_F4)  // 612 float4 slots per input
#define VCOLS     66                 // vertical-sum columns (TILE_W + 2 halo)
#define VSLOTS    (TILE_H * VCOLS)   // 2112
#define IMG_H     512
#define IMG_W     512
#define NTHREADS  256                // 8 wave32s

#if __has_builtin(__builtin_amdgcn_tensor_load_to_lds) && \
    __has_builtin(__builtin_amdgcn_s_wait_tensorcnt)
#define USE_TDM 1
#else
#define USE_TDM 0
#endif

typedef unsigned int v4u __attribute__((ext_vector_type(4)));
typedef int          v8i __attribute__((ext_vector_type(8)));
typedef int          v4i __attribute__((ext_vector_type(4)));

// Async copy 16 bytes per lane: copies the 16 bytes at global address gaddr to LDS byte offset lds_off.
// GV addressing; tracked by ASYNCcnt (CDNA5 ISA).
__device__ __forceinline__ void async_gl_to_lds_b128(uint32_t lds_off, uint64_t gaddr) {
  asm volatile("global_load_async_to_lds_b128 %0, %1, off"
               :
               : "v"(lds_off), "v"(gaddr)
               : "memory");
}

__device__ __forceinline__ void wait_async0() {
#if __has_builtin(__builtin_amdgcn_s_wait_asynccnt)
  __builtin_amdgcn_s_wait_asynccnt(0);
#else
  asm volatile("s_wait_asynccnt 0x0" ::: "memory");
#endif
}

#if USE_TDM
// One TDM descriptor: 2D tile (tile_w x tile_h elements, 4B each) from a
// 512x512 fp32 tensor (row stride 512) into packed LDS rows of tile_w floats.
// D# layout per CDNA5 ISA.
// This toolchain's builtin takes 6 args (4 descriptor groups + one extra
// 8-dword group + cache policy), per the clang-23/therock-10.0 signature.
__device__ __forceinline__ void tdm_load_2d_tile(uint32_t lds_off, const float* gptr,
                                                 int tile_w, int tile_h) {
  const uint64_t ga = (uint64_t)(uintptr_t)gptr;
  v4u g0;
  g0.x = 1u;                                         // count=1, user descriptor
  g0.y = lds_off;                                    // lds_addr (bytes)
  g0.z = (uint32_t)ga;                               // global_addr[31:0]
  g0.w = ((uint32_t)(ga >> 32) & 0x01FFFFFFu)        // global_addr[56:32]
         | 0x80000000u;                              // type=2 ("image")
  v8i g1;
  g1[0] = 0x00020000;                                // data_size=2 -> 4 bytes
  g1[1] = (int)((IMG_W & 0xFFFFu) << 16);            // tensor_dim0[15:0] @ [63:48]
  g1[2] = (int)(((IMG_W >> 16) & 0xFFFFu)            // tensor_dim0[31:16]
                | ((IMG_H & 0xFFFFu) << 16));        // tensor_dim1[15:0]
  g1[3] = (int)(((IMG_H >> 16) & 0xFFFFu)            // tensor_dim1[31:16]
                | ((uint32_t)tile_w << 16));         // tile_dim0
  g1[4] = tile_h & 0xFFFF;                           // tile_dim1 (tile_dim2=0)
  g1[5] = IMG_W;                                     // tensor_dim0_stride[31:0]
  g1[6] = 0;                                         // stride0[47:32], stride1[15:0]
  g1[7] = 0;                                         // stride1[47:16]
  v4i g2 = {0, 0, 0, 0};                             // 2D: dims 2/3 unused
  v4i g3 = {0, 0, 0, 0};
  v8i g4 = {0, 0, 0, 0, 0, 0, 0, 0};                 // extra group (unused, zero)
  __builtin_amdgcn_tensor_load_to_lds(g0, g1, g2, g3, g4, 0);
}
#endif

__global__ __launch_bounds__(NTHREADS)
void ssim_tile_kernel(const float* __restrict__ x,
                      const float* __restrict__ y,
                      float* __restrict__ out) {
  __shared__ float xs[IN_ROWS * IN_STRIDE];   // 2448 floats
  __shared__ float ys[IN_ROWS * IN_STRIDE];
  __shared__ float vsx[VSLOTS];
  __shared__ float vsy[VSLOTS];
  __shared__ float vsxx[VSLOTS];
  __shared__ float vsyy[VSLOTS];
  __shared__ float vsxy[VSLOTS];

  const int tid = threadIdx.x;
  const int bx = blockIdx.x, by = blockIdx.y, bz = blockIdx.z;
  const size_t plane = (size_t)bz * ((size_t)IMG_H * IMG_W);
  const float* xp = x + plane;
  const float* yp = y + plane;

  // Low 32 bits of a generic pointer into LDS == wave-relative LDS byte offset.
  const uint32_t xs_base = (uint32_t)(uintptr_t)(void*)xs;
  const uint32_t ys_base = (uint32_t)(uintptr_t)(void*)ys;

  const int r0 = by * TILE_H - 1;   // first staged global row
  const int c0 = bx * TILE_W - 4;   // first staged global col (16B-aligned)

  // Uniform branch: interior blocks never touch the image boundary.
  const bool interior = (bx > 0) & (bx < (IMG_W / TILE_W) - 1) &
                        (by > 0) & (by < (IMG_H / TILE_H) - 1);

  // ---- Stage 1: stage 34x72 halo tiles of x and y into LDS ----------------
  if (interior) {
#if USE_TDM
    if (tid < 32) {   // one wave issues the two DMAs for the whole block
      const size_t off = (size_t)r0 * IMG_W + c0;
      tdm_load_2d_tile(xs_base, xp + off, IN_STRIDE, IN_ROWS);
      tdm_load_2d_tile(ys_base, yp + off, IN_STRIDE, IN_ROWS);
      __builtin_amdgcn_s_wait_tensorcnt(0);
    }
#else
    for (int idx = tid; idx < IN_SLOTS4; idx += NTHREADS) {
      int r = idx / IN_F4;
      int f = idx - r * IN_F4;
      const size_t off = (size_t)(r0 + r) * IMG_W + (c0 + f * 4);
      const uint32_t loff = (uint32_t)(r * IN_STRIDE + f * 4) * 4u;
      async_gl_to_lds_b128(xs_base + loff, (uint64_t)(uintptr_t)(xp + off));
      async_gl_to_lds_b128(ys_base + loff, (uint64_t)(uintptr_t)(yp + off));
    }
#endif
  } else {
    for (int idx = tid; idx < IN_SLOTS4; idx += NTHREADS) {
      int r = idx / IN_F4;
      int f = idx - r * IN_F4;
      int gr = r0 + r;
      int gc = c0 + f * 4;          // float4 is fully in or fully out (c0 % 4 == 0)
      const uint32_t loff = (uint32_t)(r * IN_STRIDE + f * 4) * 4u;
      if ((unsigned)gr < IMG_H && (unsigned)gc < IMG_W) {
        const size_t off = (size_t)gr * IMG_W + gc;
        async_gl_to_lds_b128(xs_base + loff, (uint64_t)(uintptr_t)(xp + off));
        async_gl_to_lds_b128(ys_base + loff, (uint64_t)(uintptr_t)(yp + off));
      } else {
        // SAME padding: zero contribution for out-of-image taps.
        int s = r * IN_STRIDE + f * 4;
        xs[s + 0] = 0.0f; xs[s + 1] = 0.0f; xs[s + 2] = 0.0f; xs[s + 3] = 0.0f;
        ys[s + 0] = 0.0f; ys[s + 1] = 0.0f; ys[s + 2] = 0.0f; ys[s + 3] = 0.0f;
      }
    }
  }
  wait_async0();      // boundary path: this wave's async fills landed in LDS
  __syncthreads();    // all fills (TDM or async) visible workgroup-wide

  // ---- Stage 2: vertical 3-tap sums of the five moments -------------------
  // Output col oc uses staged cols oc+3..oc+5; vertical sums span cols 3..68.
  for (int idx = tid; idx < VSLOTS; idx += NTHREADS) {
    int r  = idx / VCOLS;
    int cv = idx - r * VCOLS;
    int b  = r * IN_STRIDE + cv + 3;
    float x0 = xs[b], x1 = xs[b + IN_STRIDE], x2 = xs[b + 2 * IN_STRIDE];
    float w0 = ys[b], w1 = ys[b + IN_STRIDE], w2 = ys[b + 2 * IN_STRIDE];
    vsx[idx]  = x0 + x1 + x2;
    vsy[idx]  = w0 + w1 + w2;
    vsxx[idx] = fmaf(x0, x0, fmaf(x1, x1, x2 * x2));
    vsyy[idx] = fmaf(w0, w0, fmaf(w1, w1, w2 * w2));
    vsxy[idx] = fmaf(x0, w0, fmaf(x1, w1, x2 * w2));
  }
  __syncthreads();

  // ---- Stage 3: horizontal 3-tap + SSIM math + coalesced store ------------
  const float C1 = 1.0e-4f;   // 0.01^2
  const float C2 = 9.0e-4f;   // 0.03^2
  for (int idx = tid; idx < TILE_H * TILE_W; idx += NTHREADS) {
    int r  = idx >> 6;                 // / TILE_W
    int oc = idx & (TILE_W - 1);
    int b  = r * VCOLS + oc;
    float sx  = vsx[b]  + vsx[b + 1]  + vsx[b + 2];
    float sy  = vsy[b]  + vsy[b + 1]  + vsy[b + 2];
    float sxx = vsxx[b] + vsxx[b + 1] + vsxx[b + 2];
    float syy = vsyy[b] + vsyy[b + 1] + vsyy[b + 2];
    float sxy = vsxy[b] + vsxy[b + 1] + vsxy[b + 2];

    int gr = by * TILE_H + r;
    int gc = bx * TILE_W + oc;
    // count_include_pad=False: separable tap counts {2,3}x{2,3};
    // reciprocals selected without division.
    float invV = (gr == 0 || gr == IMG_H - 1) ? 0.5f : (1.0f / 3.0f);
    float invH = (gc == 0 || gc == IMG_W - 1) ? 0.5f : (1.0f / 3.0f);
    float inv  = invV * invH;

    float mx  = sx  * inv, my = sy * inv;
    float ex  = sxx * inv, ey = syy * inv, exy = sxy * inv;
    float vx  = fmaf(-mx, mx, ex);
    float vy  = fmaf(-my, my, ey);
    float vxy = fmaf(-mx, my, exy);

    float num  = fmaf(2.0f * mx, my, C1) * fmaf(2.0f, vxy, C2);
    float den  = (fmaf(mx, mx, my * my) + C1) * (vx + vy + C2);
    float ssim = num / (den + 1e-12f);
    float o    = (1.0f - ssim) * 0.5f;
    o = fminf(fmaxf(o, 0.0f), 1.0f);
    out[plane + (size_t)gr * IMG_W + gc] = o;
  }
}

extern "C" void kernel_launch(void* const* d_in, const int* in_sizes, int n_in,
                              void* d_out, int out_size, void* d_ws, size_t ws_size,
                              hipStream_t stream) {
  const float* x = (const float*)d_in[0];
  const float* y = (const float*)d_in[1];
  float* out = (float*)d_out;

  const int planes = in_sizes[0] / (IMG_H * IMG_W);     // N*C = 48
  dim3 grid(IMG_W / TILE_W, IMG_H / TILE_H, planes);    // (8, 16, 48)
  dim3 block(NTHREADS);
  ssim_tile_kernel<<<grid, block, 0, stream>>>(x, y, out);
}